// OMPQuantize_62586263437526
// MI455X (gfx1250) — compile-verified
//
#include <hip/hip_runtime.h>
#include <math.h>

typedef __attribute__((ext_vector_type(2))) float v2f;
typedef __attribute__((ext_vector_type(8))) float v8f;

#define NSIG   32768
#define DIMD   64
#define KDIC   512
#define REGF   1e-7f

// d_out flat offsets (floats), tuple order:
// out[2097152], diff_enc, diff_dict, ids[16777216], num_steps, mean_D, mean_Z, norm_Z, top_pct, num_zeros
#define DIFF_OFF  2097152
#define IDS_OFF   2097154
#define SCAL_OFF  18874370

// ws float offsets
#define WS_DN   0         // 64*512
#define WS_G    32768     // 512*512
#define WS_ACC  294912    // accumulators
#define A_SSD   0
#define A_SABSD 1
#define A_SABSZ 2
#define A_HIST  3         // 5 bins (norm0 = 0..4)

__global__ void zero_kernel(float* __restrict__ out, float* __restrict__ acc) {
  // zero the dense ids region (byte offset of IDS_OFF is 8-byte aligned -> float2 ok)
  float2* ids2 = (float2*)(out + IDS_OFF);
  const size_t n2 = 16777216u / 2;
  size_t stride = (size_t)gridDim.x * blockDim.x;
  float2 z; z.x = 0.f; z.y = 0.f;
  for (size_t i = (size_t)blockIdx.x * blockDim.x + threadIdx.x; i < n2; i += stride)
    ids2[i] = z;
  if (blockIdx.x == 0 && threadIdx.x < 8) acc[threadIdx.x] = 0.f;
}

__global__ void normalize_kernel(const float* __restrict__ D, float* __restrict__ Dn,
                                 float* __restrict__ acc) {
  int k = blockIdx.x * blockDim.x + threadIdx.x;
  if (k >= KDIC) return;
  float ss = 0.f;
  for (int c = 0; c < DIMD; ++c) { float v = D[c * KDIC + k]; ss += v * v; }
  float inv = 1.f / sqrtf(ss);
  float sa = 0.f;
  for (int c = 0; c < DIMD; ++c) {
    float dn = D[c * KDIC + k] * inv;
    Dn[c * KDIC + k] = dn;
    sa += fabsf(dn);
  }
  atomicAdd(&acc[A_SABSD], sa);
}

// G = Dn^T Dn via V_WMMA_F32_16X16X4_F32; one wave per 16x16 tile of G.
__global__ void __launch_bounds__(32) gram_kernel(const float* __restrict__ Dn,
                                                  float* __restrict__ G) {
  const int l = threadIdx.x;
  const int r = l & 15, s = l >> 4;
  const int i0 = blockIdx.x * 16, j0 = blockIdx.y * 16;
  v8f acc = {};
#pragma unroll
  for (int t = 0; t < 16; ++t) {
    int c0 = 4 * t + 2 * s;
    v2f a, b;
    a.x = Dn[c0 * KDIC + i0 + r];
    a.y = Dn[(c0 + 1) * KDIC + i0 + r];
    b.x = Dn[c0 * KDIC + j0 + r];
    b.y = Dn[(c0 + 1) * KDIC + j0 + r];
    acc = __builtin_amdgcn_wmma_f32_16x16x4_f32(false, a, false, b, (short)0, acc, false, false);
  }
#pragma unroll
  for (int v = 0; v < 8; ++v)
    G[(i0 + v + 8 * s) * KDIC + j0 + r] = acc[v];
}

// Fused: per-wave 16-row tile.  alpha0 tile via WMMA -> LDS, then 4 OMP steps,
// then reconstruct/outputs.  x layout [8,64,64,64]; signal n = b*4096 + pix,
// X[n,c] = x[b*262144 + c*4096 + pix].
__global__ void __launch_bounds__(32) omp_kernel(const float* __restrict__ x,
                                                 const float* __restrict__ Dn,
                                                 const float* __restrict__ G,
                                                 float* __restrict__ out,
                                                 float* __restrict__ acc) {
  extern __shared__ float smem[];
  float* alphab = smem;                       // 16*512 f32 = 32KB (alpha0, never overwritten)
  int*   widx   = (int*)(smem + 8192);        // 16 rows x 4
  float* wgam   = (float*)(widx + 64);        // 16 rows x 4
  float* redv   = wgam + 64;                  // 32
  int*   redc   = (int*)(redv + 32);          // 32

  const int l = threadIdx.x;
  const int r = l & 15, s = l >> 4;
  const int n0 = blockIdx.x * 16;
  const int bb = n0 >> 12;
  const int pix0 = n0 & 4095;
  const size_t xbase = (size_t)bb * 262144 + (size_t)pix0;

  // Load A tile (16 rows x 64 ch): A-matrix 16x4 layout per k-step:
  // lane half s, VGPR pair holds K = 4t+2s and 4t+2s+1 for row r.
  float areg[32];
#pragma unroll
  for (int t = 0; t < 16; ++t) {
    int c0 = 4 * t + 2 * s;
    areg[2 * t]     = x[xbase + (size_t)c0 * 4096 + r];
    areg[2 * t + 1] = x[xbase + (size_t)(c0 + 1) * 4096 + r];
  }

  // alpha0[16 x 512] = A x Dn, 32 column tiles, 16 k-steps each
  for (int nt = 0; nt < 32; ++nt) {
    int nb = nt * 16;
    v8f c = {};
#pragma unroll
    for (int t = 0; t < 16; ++t) {
      int c0 = 4 * t + 2 * s;
      v2f a, b;
      a.x = areg[2 * t];
      a.y = areg[2 * t + 1];
      b.x = Dn[c0 * KDIC + nb + r];
      b.y = Dn[(c0 + 1) * KDIC + nb + r];
      c = __builtin_amdgcn_wmma_f32_16x16x4_f32(false, a, false, b, (short)0, c, false, false);
    }
#pragma unroll
    for (int v = 0; v < 8; ++v)
      alphab[(v + 8 * s) * KDIC + nb + r] = c[v];
  }
  __syncthreads();

  // ---- 4 OMP steps (k unrolled -> constant inner bounds, registers stay registers)
#pragma unroll
  for (int k = 0; k < 4; ++k) {
    int   id[4]; float gm[4];
    for (int j = 0; j < k; ++j) { id[j] = widx[r * 4 + j]; gm[j] = wgam[r * 4 + j]; }
    // argmax |alpha0 - G[idx]^T gamma| ; 2 lanes per row (halves of 512 cols)
    float best = -1.f; int bestc = 0;
    const int cb = s * 256;
    for (int cc = 0; cc < 256; ++cc) {
      int c = cb + cc;
      float v = alphab[r * KDIC + c];
      for (int j = 0; j < k; ++j) v -= gm[j] * G[(size_t)id[j] * KDIC + c];
      float av = fabsf(v);
      if (av > best) { best = av; bestc = c; }   // strict > keeps first index
    }
    redv[l] = best; redc[l] = bestc;
    __syncthreads();
    if (l < 16) {
      // cross-half combine: tie -> lower column (half 0), matching jnp.argmax
      float v1 = redv[l + 16];
      int bc = (v1 > best) ? redc[l + 16] : bestc;
      widx[l * 4 + k] = bc;
      id[k] = bc;
      // build (k+1)x(k+1) G_sub + reg*I, rhs = alpha0[row, idx]; solve (SPD, no pivot)
      float Gm[4][4], rhs[4], gg[4];
      for (int a = 0; a <= k; ++a) {
        rhs[a] = alphab[l * KDIC + id[a]];
        for (int b2 = 0; b2 <= k; ++b2)
          Gm[a][b2] = G[(size_t)id[a] * KDIC + id[b2]] + ((a == b2) ? REGF : 0.f);
      }
      for (int p = 0; p <= k; ++p) {
        float inv = 1.f / Gm[p][p];
        for (int rr = p + 1; rr <= k; ++rr) {
          float f = Gm[rr][p] * inv;
          for (int c2 = p; c2 <= k; ++c2) Gm[rr][c2] -= f * Gm[p][c2];
          rhs[rr] -= f * rhs[p];
        }
      }
      for (int p = k; p >= 0; --p) {
        float v = rhs[p];
        for (int c2 = p + 1; c2 <= k; ++c2) v -= Gm[p][c2] * gg[c2];
        gg[p] = v / Gm[p][p];
      }
      for (int j = 0; j <= k; ++j) wgam[l * 4 + j] = gg[j];
    }
    __syncthreads();
  }

  // ---- outputs
  int   id[4]; float gm[4];
#pragma unroll
  for (int j = 0; j < 4; ++j) { id[j] = widx[r * 4 + j]; gm[j] = wgam[r * 4 + j]; }

  // quant -> out (same [B,64,H,W] layout as x), accumulate squared error
  float lss = 0.f;
  for (int cc = 0; cc < 32; ++cc) {
    int c = s * 32 + cc;
    float q = 0.f;
#pragma unroll
    for (int j = 0; j < 4; ++j) q += gm[j] * Dn[c * KDIC + id[j]];
    size_t off = (size_t)bb * 262144 + (size_t)c * 4096 + pix0 + r;
    out[off] = q;
    float d = q - x[off];
    lss += d * d;
  }
  atomicAdd(&acc[A_SSD], lss);

  if (l < 16) {
    float* ids = out + IDS_OFF;  // [B,512,64,64]
    float sAbs = 0.f; int cnt = 0;
#pragma unroll
    for (int j = 0; j < 4; ++j) {
      bool last = true;
#pragma unroll
      for (int j2 = j + 1; j2 < 4; ++j2) if (id[j2] == id[j]) last = false;
      if (last && gm[j] != 0.f) { cnt++; sAbs += fabsf(gm[j]); }
      ids[(size_t)bb * 2097152 + (size_t)id[j] * 4096 + pix0 + r] = gm[j];
    }
    atomicAdd(&acc[A_SABSZ], sAbs);
    atomicAdd(&acc[A_HIST + cnt], 1.f);
  }
}

__global__ void finalize_kernel(const float* __restrict__ acc, float* __restrict__ out) {
  if (threadIdx.x != 0 || blockIdx.x != 0) return;
  float diff = acc[A_SSD] / 2097152.f;        // mean over 32768*64
  out[DIFF_OFF]     = diff;
  out[DIFF_OFF + 1] = diff;
  float* sc = out + SCAL_OFF;
  sc[0] = 4.f;                                 // num_steps
  sc[1] = acc[A_SABSD] / 32768.f;              // mean|Dn| over 64*512
  sc[2] = acc[A_SABSZ] / 16777216.f;           // mean|code| over 32768*512
  float sumN = 0.f;
  for (int v = 0; v < 5; ++v) sumN += (float)v * acc[A_HIST + v];
  sc[3] = sumN / 32768.f;                      // norm_Z
  float cum = 0.f, top = 0.f;                  // min of top-327 norm0 values
  for (int v = 4; v >= 0; --v) { cum += acc[A_HIST + v]; if (cum >= 327.f) { top = (float)v; break; } }
  sc[4] = top;
  sc[5] = acc[A_HIST + 0];                     // num_zeros
}

extern "C" void kernel_launch(void* const* d_in, const int* in_sizes, int n_in,
                              void* d_out, int out_size, void* d_ws, size_t ws_size,
                              hipStream_t stream) {
  (void)in_sizes; (void)n_in; (void)out_size; (void)ws_size;
  const float* x = (const float*)d_in[0];
  const float* D = (const float*)d_in[1];
  float* out = (float*)d_out;
  float* ws  = (float*)d_ws;
  float* Dn  = ws + WS_DN;
  float* G   = ws + WS_G;
  float* acc = ws + WS_ACC;

  zero_kernel<<<8192, 256, 0, stream>>>(out, acc);
  normalize_kernel<<<2, 256, 0, stream>>>(D, Dn, acc);
  gram_kernel<<<dim3(32, 32), 32, 0, stream>>>(Dn, G);
  size_t shmem = (8192 + 64 + 64 + 32 + 32) * sizeof(float) + 64;
  omp_kernel<<<NSIG / 16, 32, shmem, stream>>>(x, Dn, G, out, acc);
  finalize_kernel<<<1, 1, 0, stream>>>(acc, out);
}